// MQIFNeuronTorch_24352464569939
// MI455X (gfx1250) — compile-verified
//
#include <hip/hip_runtime.h>
#include <stdint.h>

// MQIF scan: batch=16, steps=4096, features=512.
// d_out = [ v_trace (16*4097*512 f32) | spikes (16*4097*512 f32) ]

#define BATCH 16
#define STEPS 4096
#define FEAT  512
#define FPB   128                 // features per block (4 wave32 waves)
#define CH    32                  // time steps per staged LDS tile
#define NCH   (STEPS / CH)        // 128 chunks
#define NBUF  4                   // LDS ring depth (4 x 16KB = 64KB)
#define CHBYTES (CH * FPB * 4)    // 16384 bytes per tile

// ---- CDNA5 TDM availability (device pass only; host pass sees fallback) ----
#if defined(__HIP_DEVICE_COMPILE__) && __has_builtin(__builtin_amdgcn_tensor_load_to_lds)
#define HAVE_TDM 1
#else
#define HAVE_TDM 0
#endif

#if HAVE_TDM
typedef unsigned int v4u __attribute__((ext_vector_type(4)));
typedef int          v4i __attribute__((ext_vector_type(4)));
typedef int          v8i __attribute__((ext_vector_type(8)));

// Issue one TDM load of a CH x FPB f32 tile (row stride FEAT) into LDS.
// D# packed per CDNA5 ISA 8.3/8.4: 2D tile, data_size=4B, no padding,
// no multicast (workgroup_mask=0), dims sized so no OOB path is taken.
__device__ __forceinline__ void tdm_load_tile(const void* gsrc, unsigned lds_off) {
  unsigned long long ga = (unsigned long long)(uintptr_t)gsrc;
  v4u g0;
  g0[0] = 1u;                                              // count=1, user mode
  g0[1] = lds_off;                                         // LDS byte address
  g0[2] = (unsigned)(ga & 0xFFFFFFFFull);                  // global_addr[31:0]
  g0[3] = ((unsigned)(ga >> 32) & 0x01FFFFFFu) | (2u << 30); // addr[56:32] | type=2
  v8i g1;
  g1[0] = 0x00020000;                        // workgroup_mask=0, data_size=2 (4B)
  g1[1] = (int)((unsigned)FEAT  << 16);      // tensor_dim0 lo16 = 512
  g1[2] = (int)((unsigned)STEPS << 16);      // tensor_dim0 hi16=0 | tensor_dim1 lo16 = 4096
  g1[3] = (int)((unsigned)FPB   << 16);      // tensor_dim1 hi16=0 | tile_dim0 = 128
  g1[4] = CH;                                // tile_dim1 = 32, tile_dim2 = 0
  g1[5] = FEAT;                              // tensor_dim0_stride lo32 = 512 elems
  g1[6] = 0;                                 // stride0 hi | stride1 lo (unused, 2D)
  g1[7] = 0;
  v4i gz = {0, 0, 0, 0};
#if defined(__clang_major__) && (__clang_major__ >= 23)
  v8i gz8 = {0, 0, 0, 0, 0, 0, 0, 0};
  __builtin_amdgcn_tensor_load_to_lds(g0, g1, gz, gz, gz8, 0);
#else
  __builtin_amdgcn_tensor_load_to_lds(g0, g1, gz, gz, 0);
#endif
}
#endif // HAVE_TDM

// One MQIF step. Short dependent chain:
//   a0,a1 (d1) -> m (d2) -> v' = fma(0.0002,m,pre) (d3) -> select (d4)
// pre = v + 0.005*(i - u) computed off the critical path.
__device__ __forceinline__ void mqif_step(const float i_t, float& v, float& u,
                                          float*& pv, float*& ps) {
  const bool fired = (v >= 30.0f);
  __builtin_nontemporal_store(fired ? 30.0f : v, pv);     // clamped visible v
  __builtin_nontemporal_store(fired ? 1.0f : 0.0f, ps);   // spike flag
  const float a0  = v + 60.0f;                            // v - VR
  const float a1  = v + 40.0f;                            // v - VT
  const float m   = a0 * a1;
  const float pre = __builtin_fmaf(0.005f, i_t - u, v);   // v + DT/TAU_M*(i-u)
  const float vn  = __builtin_fmaf(0.0002f, m, pre);      // + DT/TAU_M*A*m
  const float un  = __builtin_fmaf(0.0005f,
                      __builtin_fmaf(0.2f, a0, -u), u);   // u + DT/TAU_U*(B*a0-u)
  const float uj  = u + 2.0f;                             // u + D on fire
  v = fired ? -60.0f : vn;                                // V_RESET on fire
  u = fired ? uj : un;
  pv += FEAT;
  ps += FEAT;
}

__global__ void __launch_bounds__(FPB, 1)
mqif_scan_kernel(const float* __restrict__ in, float* __restrict__ vtr,
                 float* __restrict__ spk) {
  const int tid = (int)threadIdx.x;
  const int b   = (int)blockIdx.x >> 2;            // 16 batches
  const int f0  = ((int)blockIdx.x & 3) * FPB;     // 4 feature quarters
  const int f   = f0 + tid;

  float* pv = vtr + (size_t)b * (STEPS + 1) * FEAT + f;
  float* ps = spk + (size_t)b * (STEPS + 1) * FEAT + f;

  float v = -60.0f;   // VR
  float u = 0.0f;

#if HAVE_TDM
  __shared__ float buf[NBUF][CH * FPB];            // 4 x 16KB ring buffer
  const float* gtile = in + (size_t)b * STEPS * FEAT + f0;
  // Generic->LDS aperture: low 32 bits of a shared pointer are the LDS address.
  const unsigned lds_base = (unsigned)(uintptr_t)&buf[0][0];

  // Prologue: fill pipeline 3 deep (chunks 0..2 -> buffers 0..2).
  if (tid < 32) {
    tdm_load_tile(gtile,                       lds_base);
    tdm_load_tile(gtile + (size_t)1 * CH * FEAT, lds_base + 1 * CHBYTES);
    tdm_load_tile(gtile + (size_t)2 * CH * FEAT, lds_base + 2 * CHBYTES);
  }

  for (int c = 0; c < NCH; ++c) {
    if (tid < 32) {
      // Issue chunk c+3 into buffer (c+3)&3 == (c-1)&3: its readers finished
      // at the closing barrier of iteration c-1, before this point.
      if (c + 3 < NCH) {
        tdm_load_tile(gtile + (size_t)(c + 3) * CH * FEAT,
                      lds_base + (unsigned)((c + 3) & (NBUF - 1)) * CHBYTES);
        __builtin_amdgcn_s_wait_tensorcnt(3);    // in-order: chunk c complete
      } else if (c + 2 < NCH) {
        __builtin_amdgcn_s_wait_tensorcnt(2);
      } else if (c + 1 < NCH) {
        __builtin_amdgcn_s_wait_tensorcnt(1);
      } else {
        __builtin_amdgcn_s_wait_tensorcnt(0);
      }
    }
    __syncthreads();                             // tile c visible to all waves
    const float* lb = &buf[0][0] + (size_t)(c & (NBUF - 1)) * (CH * FPB);
#pragma unroll 8
    for (int s = 0; s < CH; ++s) {
      const float i_t = lb[s * FPB + tid];       // conflict-free ds_load_b32
      mqif_step(i_t, v, u, pv, ps);
    }
    __syncthreads();                             // done reading before refill
  }
#else
  // Fallback: direct streamed loads + explicit prefetch ahead of the scan.
  const float* gp = in + (size_t)b * STEPS * FEAT + f;
  for (int t = 0; t < STEPS; ++t) {
    __builtin_prefetch(gp + 64 * FEAT, 0, 0);    // global_prefetch_b8, 128KB ahead
    const float i_t = __builtin_nontemporal_load(gp);
    gp += FEAT;
    mqif_step(i_t, v, u, pv, ps);
  }
#endif

  // Row STEPS: raw final voltage and final fired flag.
  __builtin_nontemporal_store(v, pv);
  __builtin_nontemporal_store((v >= 30.0f) ? 1.0f : 0.0f, ps);
}

extern "C" void kernel_launch(void* const* d_in, const int* in_sizes, int n_in,
                              void* d_out, int out_size, void* d_ws, size_t ws_size,
                              hipStream_t stream) {
  (void)in_sizes; (void)n_in; (void)out_size; (void)d_ws; (void)ws_size;
  const float* in = (const float*)d_in[0];
  float* vtr = (float*)d_out;
  float* spk = vtr + (size_t)BATCH * (STEPS + 1) * FEAT;
  dim3 grid(BATCH * (FEAT / FPB));   // 64 blocks
  dim3 block(FPB);                   // 128 threads = 4 wave32 waves
  mqif_scan_kernel<<<grid, block, 0, stream>>>(in, vtr, spk);
}